// net_39041252721195
// MI455X (gfx1250) — compile-verified
//
#include <hip/hip_runtime.h>
#include <math.h>

typedef __attribute__((ext_vector_type(2))) float v2f;
typedef __attribute__((ext_vector_type(8))) float v8f;

#if defined(__gfx1250__) && __has_builtin(__builtin_amdgcn_wmma_f32_16x16x4_f32)
#define USE_WMMA 1
#else
#define USE_WMMA 0
#endif

#define EPB 256           // edges per block in edge pass 2
#define NEG_INF (-__builtin_inff())

__device__ __forceinline__ int iclamp01(int v) { return v < 0 ? 0 : (v > 1 ? 1 : v); }

// float atomic max via int/uint ordering trick (works for global and LDS ptrs)
__device__ __forceinline__ void atomicMaxF(float* addr, float v) {
    if (v >= 0.0f) atomicMax((int*)addr, __float_as_int(v));
    else           atomicMin((unsigned int*)addr, __float_as_uint(v));
}

// ---------------------------------------------------------------- init ws
__global__ void k_init(float* ws, long long nzero, long long total) {
    long long i = (long long)blockIdx.x * blockDim.x + threadIdx.x;
    long long gs = (long long)gridDim.x * blockDim.x;
    for (; i < total; i += gs)
        ws[i] = (i < nzero) ? 0.0f : NEG_INF;   // tail region = pooled cells -> -inf
}

// ------------------------------------------------- spline-conv layer 1 (in=1,out=2)
__global__ void k_edge1(const float* __restrict__ x, const int* __restrict__ ei,
                        long long E, const float* __restrict__ ea,
                        const float* __restrict__ W1,
                        float* __restrict__ sum1, float* __restrict__ cnt) {
    __shared__ float sW[16];
    if (threadIdx.x < 16) sW[threadIdx.x] = W1[threadIdx.x];
    __syncthreads();
    long long i = (long long)blockIdx.x * blockDim.x + threadIdx.x;
    long long gs = (long long)gridDim.x * blockDim.x;
    for (long long e = i; e < E; e += gs) {
        int s = ei[e], d = ei[E + e];
        float p0 = ea[3*e], p1 = ea[3*e+1], p2 = ea[3*e+2];
        int l0 = (int)floorf(p0), l1 = (int)floorf(p1), l2 = (int)floorf(p2);
        float f0 = p0 - (float)l0, f1 = p1 - (float)l1, f2 = p2 - (float)l2;
        float xs = x[s];
        float c0 = 0.f, c1 = 0.f;
#pragma unroll
        for (int b = 0; b < 8; ++b) {
            int b0 = b & 1, b1 = (b >> 1) & 1, b2 = (b >> 2) & 1;
            float w = (b0 ? f0 : 1.f - f0) * (b1 ? f1 : 1.f - f1) * (b2 ? f2 : 1.f - f2);
            int idx = iclamp01(l0 + b0) + 2 * iclamp01(l1 + b1) + 4 * iclamp01(l2 + b2);
            c0 += w * sW[idx * 2 + 0];
            c1 += w * sW[idx * 2 + 1];
        }
        atomicAdd(&sum1[2*d + 0], xs * c0);
        atomicAdd(&sum1[2*d + 1], xs * c1);
        atomicAdd(&cnt[d], 1.0f);
    }
}

// ------------------------------ mean + ELU + BN1 stats (sum / sumsq per channel)
__global__ void k_node1(const float* __restrict__ sum1, const float* __restrict__ cnt,
                        float* __restrict__ h1, float* __restrict__ stats1, int N) {
    int i = blockIdx.x * blockDim.x + threadIdx.x;
    int gs = gridDim.x * blockDim.x;
    float s0 = 0.f, s1 = 0.f, q0 = 0.f, q1 = 0.f;
    for (int n = i; n < N; n += gs) {
        float inv = 1.0f / fmaxf(cnt[n], 1.0f);
        float a = sum1[2*n] * inv, b = sum1[2*n+1] * inv;
        a = a > 0.f ? a : expm1f(a);
        b = b > 0.f ? b : expm1f(b);
        h1[2*n] = a; h1[2*n+1] = b;
        s0 += a; s1 += b; q0 += a*a; q1 += b*b;
    }
#pragma unroll
    for (int off = 16; off > 0; off >>= 1) {
        s0 += __shfl_down(s0, off); s1 += __shfl_down(s1, off);
        q0 += __shfl_down(q0, off); q1 += __shfl_down(q1, off);
    }
    if ((threadIdx.x & 31) == 0) {
        atomicAdd(&stats1[0], s0); atomicAdd(&stats1[1], s1);
        atomicAdd(&stats1[2], q0); atomicAdd(&stats1[3], q1);
    }
}

__global__ void k_fin1(const float* __restrict__ stats1, const float* __restrict__ g,
                       const float* __restrict__ b, float* __restrict__ scale1, float fN) {
    int t = threadIdx.x;
    if (t < 2) {
        float mu  = stats1[t] / fN;
        float var = stats1[2 + t] / fN - mu * mu;
        float a   = rsqrtf(var + 1e-5f) * g[t];
        scale1[t]     = a;
        scale1[2 + t] = b[t] - mu * a;
    }
}

// ---------------- spline-conv layer 2 as [E,16]x[16,4] GEMM via V_WMMA_F32_16X16X4_F32
__global__ void __launch_bounds__(EPB)
k_edge2(const float* __restrict__ h1, const int* __restrict__ ei, long long E,
        const float* __restrict__ ea, const float* __restrict__ W2,
        const float* __restrict__ scale1, float* __restrict__ sum2) {
    __shared__ __align__(16) float uBuf[EPB * 18];   // per-edge u[16], stride 18
    __shared__ int   dBuf[EPB];
    __shared__ float oBuf[EPB * 4];

    int tid = threadIdx.x;
    long long e = (long long)blockIdx.x * EPB + tid;
    bool valid = e < E;
    long long ec = valid ? e : (E - 1);

    int s = ei[ec], d = ei[E + ec];
    float a0 = scale1[0], a1 = scale1[1], c0 = scale1[2], c1 = scale1[3];
    float hs0 = h1[2*s]   * a0 + c0;          // BN1 applied on the fly
    float hs1 = h1[2*s+1] * a1 + c1;

    float p0 = ea[3*ec], p1 = ea[3*ec+1], p2 = ea[3*ec+2];
    int l0 = (int)floorf(p0), l1 = (int)floorf(p1), l2 = (int)floorf(p2);
    float f0 = p0 - (float)l0, f1 = p1 - (float)l1, f2 = p2 - (float)l2;

    float* u = &uBuf[tid * 18];
#pragma unroll
    for (int k = 0; k < 16; ++k) u[k] = 0.f;
    if (valid) {
#pragma unroll
        for (int b = 0; b < 8; ++b) {
            int b0 = b & 1, b1 = (b >> 1) & 1, b2 = (b >> 2) & 1;
            float w = (b0 ? f0 : 1.f - f0) * (b1 ? f1 : 1.f - f1) * (b2 ? f2 : 1.f - f2);
            int idx = iclamp01(l0 + b0) + 2 * iclamp01(l1 + b1) + 4 * iclamp01(l2 + b2);
            u[2*idx + 0] += w * hs0;          // u[k]=w_b*h_i with k=2*idx+i
            u[2*idx + 1] += w * hs1;
        }
    }
    dBuf[tid] = valid ? d : -1;
    __syncthreads();

    int lane = tid & 31, wv = tid >> 5;
#if USE_WMMA
    // B (4x16 f32, chunks c=0..3): lane n=lane&15 cols, K split across half-waves.
    int ncol = lane & 15, khalf = lane >> 4;
    v2f Breg[4];
#pragma unroll
    for (int c = 0; c < 4; ++c) {
        int k0 = 4 * c + 2 * khalf;
        float bx = 0.f, by = 0.f;
        if (ncol < 4) {                       // Wflat[k][o] = W2[k>>1][k&1][o]
            bx = W2[(k0 >> 1) * 8 + (k0 & 1) * 4 + ncol];
            by = W2[((k0 + 1) >> 1) * 8 + ((k0 + 1) & 1) * 4 + ncol];
        }
        Breg[c] = (v2f){bx, by};
    }
#pragma unroll
    for (int t = 0; t < 2; ++t) {            // each wave: 2 tiles of 16 edges
        int tb = (2 * wv + t) * 16;
        v8f acc = {};
#pragma unroll
        for (int c = 0; c < 4; ++c) {        // chain K=4 chunks -> K=16
            int m  = lane & 15;
            int kb = 4 * c + 2 * (lane >> 4);
            v2f a = *(const v2f*)&uBuf[(tb + m) * 18 + kb];
            acc = __builtin_amdgcn_wmma_f32_16x16x4_f32(
                false, a, false, Breg[c], (short)0, acc, false, false);
        }
        int col = lane & 15, half = lane >> 4;
        if (col < 4) {                       // D(m,n): VGPR r holds M=r / M=8+r
#pragma unroll
            for (int r = 0; r < 8; ++r)
                oBuf[(tb + half * 8 + r) * 4 + col] = acc[r];
        }
    }
#else
    // Scalar fallback: msg[o] = sum_k u[k] * Wflat[k][o]
#pragma unroll
    for (int o = 0; o < 4; ++o) {
        float acc = 0.f;
#pragma unroll
        for (int k = 0; k < 16; ++k)
            acc += u[k] * W2[(k >> 1) * 8 + (k & 1) * 4 + o];
        oBuf[tid * 4 + o] = acc;
    }
    (void)lane; (void)wv;
#endif
    __syncthreads();
    int dd = dBuf[tid];
    if (dd >= 0) {
#pragma unroll
        for (int o = 0; o < 4; ++o)
            atomicAdd(&sum2[4 * dd + o], oBuf[tid * 4 + o]);
    }
}

// ------------------------------ mean + BN2 stats
__global__ void k_node2(const float* __restrict__ sum2, const float* __restrict__ cnt,
                        float* __restrict__ h2, float* __restrict__ stats2, int N) {
    int i = blockIdx.x * blockDim.x + threadIdx.x;
    int gs = gridDim.x * blockDim.x;
    float s[4] = {0,0,0,0}, q[4] = {0,0,0,0};
    for (int n = i; n < N; n += gs) {
        float inv = 1.0f / fmaxf(cnt[n], 1.0f);
#pragma unroll
        for (int o = 0; o < 4; ++o) {
            float v = sum2[4*n + o] * inv;
            h2[4*n + o] = v;
            s[o] += v; q[o] += v * v;
        }
    }
#pragma unroll
    for (int off = 16; off > 0; off >>= 1) {
#pragma unroll
        for (int o = 0; o < 4; ++o) {
            s[o] += __shfl_down(s[o], off);
            q[o] += __shfl_down(q[o], off);
        }
    }
    if ((threadIdx.x & 31) == 0) {
#pragma unroll
        for (int o = 0; o < 4; ++o) {
            atomicAdd(&stats2[o], s[o]);
            atomicAdd(&stats2[4 + o], q[o]);
        }
    }
}

__global__ void k_fin2(const float* __restrict__ stats2, const float* __restrict__ g,
                       const float* __restrict__ b, float* __restrict__ scale2, float fN) {
    int t = threadIdx.x;
    if (t < 4) {
        float mu  = stats2[t] / fN;
        float var = stats2[4 + t] / fN - mu * mu;
        float a   = rsqrtf(var + 1e-5f) * g[t];
        scale2[t]     = a;
        scale2[4 + t] = b[t] - mu * a;
    }
}

// ------------------------------ 4x4 grid max pool (BN2 applied on the fly)
__global__ void k_pool(const float* __restrict__ h2, const float* __restrict__ pos,
                       const float* __restrict__ scale2, float* __restrict__ pooled, int N) {
    __shared__ float sP[64];
    if (threadIdx.x < 64) sP[threadIdx.x] = NEG_INF;
    __syncthreads();
    int i = blockIdx.x * blockDim.x + threadIdx.x;
    int gs = gridDim.x * blockDim.x;
    for (int n = i; n < N; n += gs) {
        float px = pos[2*n], py = pos[2*n+1];
        int cx = (int)floorf(px * 0.04f);  cx = cx < 0 ? 0 : (cx > 3 ? 3 : cx);
        int cy = (int)floorf(py * 0.04f);  cy = cy < 0 ? 0 : (cy > 3 ? 3 : cy);
        int cl = cx + 4 * cy;
#pragma unroll
        for (int o = 0; o < 4; ++o) {
            float v = h2[4*n + o] * scale2[o] + scale2[4 + o];
            atomicMaxF(&sP[cl * 4 + o], v);
        }
    }
    __syncthreads();
    if (threadIdx.x < 64) atomicMaxF(&pooled[threadIdx.x], sP[threadIdx.x]);
}

// ------------------------------ final FC: [1,64] @ [64,4]
__global__ void k_fc(const float* __restrict__ pooled, const float* __restrict__ fcw,
                     float* __restrict__ out) {
    __shared__ float flat[64];
    int t = threadIdx.x;
    if (t < 64) {
        float v = pooled[t];
        unsigned bits = __float_as_uint(v);
        flat[t] = ((bits & 0x7f800000u) == 0x7f800000u) ? 0.0f : v;  // !isfinite -> 0
    }
    __syncthreads();
    if (t < 4) {
        float acc = 0.f;
#pragma unroll
        for (int k = 0; k < 64; ++k) acc += flat[k] * fcw[t * 64 + k];
        out[t] = acc;
    }
}

extern "C" void kernel_launch(void* const* d_in, const int* in_sizes, int n_in,
                              void* d_out, int out_size, void* d_ws, size_t ws_size,
                              hipStream_t stream) {
    const float* x   = (const float*)d_in[0];
    const int*   ei  = (const int*)d_in[1];     // [2,E] int32
    const float* ea  = (const float*)d_in[2];   // [E,3]
    const float* pos = (const float*)d_in[3];   // [N,2]
    const float* W1  = (const float*)d_in[4];   // [8,1,2]
    const float* W2  = (const float*)d_in[5];   // [8,2,4]
    const float* g1  = (const float*)d_in[6];
    const float* b1  = (const float*)d_in[7];
    const float* g2  = (const float*)d_in[8];
    const float* b2  = (const float*)d_in[9];
    const float* fcw = (const float*)d_in[10];  // [4,64]
    float* out = (float*)d_out;

    long long N = in_sizes[0];
    long long E = (long long)in_sizes[2] / 3;

    float* ws     = (float*)d_ws;
    float* sum1   = ws;                 // 2N
    float* cnt    = ws + 2 * N;         // N
    float* h1     = ws + 3 * N;         // 2N
    float* sum2   = ws + 5 * N;         // 4N
    float* h2     = ws + 9 * N;         // 4N
    float* stats1 = ws + 13 * N;        // 4
    float* stats2 = stats1 + 4;         // 8
    float* scale1 = stats1 + 12;        // 4
    float* scale2 = stats1 + 16;        // 8
    float* pooled = stats1 + 24;        // 64
    long long nzero = 13 * N + 24;
    long long total = 13 * N + 88;

    k_init<<<1024, 256, 0, stream>>>(ws, nzero, total);

    k_edge1<<<8192, 256, 0, stream>>>(x, ei, E, ea, W1, sum1, cnt);
    k_node1<<<2048, 256, 0, stream>>>(sum1, cnt, h1, stats1, (int)N);
    k_fin1<<<1, 32, 0, stream>>>(stats1, g1, b1, scale1, (float)N);

    int eb2 = (int)((E + EPB - 1) / EPB);
    k_edge2<<<eb2, EPB, 0, stream>>>(h1, ei, E, ea, W2, scale1, sum2);
    k_node2<<<2048, 256, 0, stream>>>(sum2, cnt, h2, stats2, (int)N);
    k_fin2<<<1, 32, 0, stream>>>(stats2, g2, b2, scale2, (float)N);

    k_pool<<<2048, 256, 0, stream>>>(h2, pos, scale2, pooled, (int)N);
    k_fc<<<1, 64, 0, stream>>>(pooled, fcw, out);
}